// E60ResidualNonlinearCell_45792941310471
// MI455X (gfx1250) — compile-verified
//
#include <hip/hip_runtime.h>
#include <hip/hip_bf16.h>

typedef __attribute__((ext_vector_type(2))) float v2f;
typedef __attribute__((ext_vector_type(4))) float v4f;
typedef __attribute__((ext_vector_type(8))) float v8f;

#define T_STEPS 2048
#define BATCH   16
#define DIM     1024
#define NGROUPS (DIM / 16)     // 64 K-groups of 16
#define SCAN_WGS   16
#define SCAN_WAVES 4

__device__ __forceinline__ v8f wmma_f32(v2f a, v2f b, v8f c) {
  // D = A(16x4 f32) * B(4x16 f32) + C(16x16 f32), full fp32 precision
  return __builtin_amdgcn_wmma_f32_16x16x4_f32(false, a, false, b,
                                               (short)0, c, false, false);
}

// ---------------------------------------------------------------------------
// Stage a 16 x 1024 f32 operand tile into LDS in WMMA A-fragment order:
//   AF[g][lane][j] , g = k/16 group (64), lane = 0..31, j = 0..7 floats
//   lane (lo,hi): holds rows lo, K offsets {0,1,4,5,8,9,12,13} + 2*hi
// A lane's data for one 16-K group is 32 contiguous bytes ->
// the compute loop reads it with ds_load_b128 at (lane*32 + g*1024).
// ---------------------------------------------------------------------------
__device__ __forceinline__ void stage_frags(const float* __restrict__ src,
                                            float* __restrict__ lds,
                                            int nthreads) {
  const v4f* s4 = (const v4f*)src;     // 16 rows x 256 float4
  for (int s = threadIdx.x; s < 4096; s += nthreads) {
    int g   = s >> 6;                  // K group
    int row = (s >> 2) & 15;
    int u   = s & 3;                   // float4 within group: k = 16g + 4u
    v4f v = s4[row * 256 + (g << 2) + u];
    int base = (g << 8) + (row << 3) + (u << 1);
    v2f pl; pl[0] = v[0]; pl[1] = v[1];   // K offsets 4u+0,1 -> hi=0 lanes
    v2f ph; ph[0] = v[2]; ph[1] = v[3];   // K offsets 4u+2,3 -> hi=1 lanes
    *(v2f*)(lds + base)       = pl;
    *(v2f*)(lds + base + 128) = ph;       // lane row+16
  }
}

// Recover element (row, col) of the staged tile from fragment order.
__device__ __forceinline__ float lds_elem(const float* __restrict__ lds,
                                          int row, int col) {
  int l = row + ((col & 2) << 3);          // +16 lanes hold k%4 in {2,3}
  int j = ((col >> 1) & 6) + (col & 1);    // 2*((col>>2)&3) + (col&1)
  return lds[((col >> 4) << 8) + (l << 3) + j];
}

// K-loop: C += A(staged) * B(row 'col' of W), 4 independent WMMA chains.
__device__ __forceinline__ void gemm_k(const float* __restrict__ afrag_base,
                                       const float* __restrict__ wrow, int d2,
                                       v8f& c0, v8f& c1, v8f& c2, v8f& c3) {
#pragma unroll 4
  for (int g = 0; g < NGROUPS; ++g) {
    const float* ap = afrag_base + (g << 8);   // single base + imm offsets
    v4f q0 = *(const v4f*)(ap);                // frags 0,1  (ds_load_b128)
    v4f q1 = *(const v4f*)(ap + 4);            // frags 2,3
    const float* wp = wrow + (g << 4) + d2;
    v2f b0 = *(const v2f*)(wp + 0);
    v2f b1 = *(const v2f*)(wp + 4);
    v2f b2 = *(const v2f*)(wp + 8);
    v2f b3 = *(const v2f*)(wp + 12);
    v2f a0 = __builtin_shufflevector(q0, q0, 0, 1);
    v2f a1 = __builtin_shufflevector(q0, q0, 2, 3);
    v2f a2 = __builtin_shufflevector(q1, q1, 0, 1);
    v2f a3 = __builtin_shufflevector(q1, q1, 2, 3);
    c0 = wmma_f32(a0, b0, c0);
    c1 = wmma_f32(a1, b1, c1);
    c2 = wmma_f32(a2, b2, c2);
    c3 = wmma_f32(a3, b3, c3);
  }
}

// ---------------------------------------------------------------------------
// Phase 1: outs[t,b,e] = sum_d x[t,b,d] * W_x[e,d] + bias[e]
// ---------------------------------------------------------------------------
__global__ __launch_bounds__(256) void e60_wx_gemm(
    const float* __restrict__ x, const float* __restrict__ Wx,
    const float* __restrict__ bias, float* __restrict__ outs,
    unsigned* __restrict__ bar)
{
  if (blockIdx.x == 0 && blockIdx.y == 0 && threadIdx.x == 0) *bar = 0u;

  __shared__ float lds[BATCH * DIM];
  const int lane = threadIdx.x & 31;
  const int wave = threadIdx.x >> 5;
  const int lo   = lane & 15;
  const int hi   = lane >> 4;
  const int mtile = blockIdx.x;

  stage_frags(x + (size_t)mtile * BATCH * DIM, lds, 256);
  __syncthreads();

  const int col = (blockIdx.y * 8 + wave) * 16 + lo;   // output feature e
  const float* wrow = Wx + (size_t)col * DIM;          // B[k][col] = W_x[col][k]

  v8f c0, c1, c2, c3;
  const float bv = bias[col];
#pragma unroll
  for (int i = 0; i < 8; ++i) { c0[i] = bv; c1[i] = 0.f; c2[i] = 0.f; c3[i] = 0.f; }

  gemm_k(lds + (lane << 3), wrow, 2 * hi, c0, c1, c2, c3);

  float* ob = outs + (size_t)mtile * BATCH * DIM;
#pragma unroll
  for (int i = 0; i < 8; ++i) {
    float r = (c0[i] + c1[i]) + (c2[i] + c3[i]);
    ob[(size_t)(i + 8 * hi) * DIM + col] = r;
  }
}

// ---------------------------------------------------------------------------
// Phase 2: sequential scan; 16 WGs x 4 waves, one 16x16 tile per wave.
// ---------------------------------------------------------------------------
__global__ __launch_bounds__(128) void e60_scan(
    const float* __restrict__ Wh, const float* __restrict__ log_alpha_p,
    float* __restrict__ outs, float* __restrict__ h,
    volatile unsigned* __restrict__ bar)
{
  __shared__ float lds[BATCH * DIM];
  const int lane = threadIdx.x & 31;
  const int wave = threadIdx.x >> 5;
  const int lo   = lane & 15;
  const int hi   = lane >> 4;

  const int col = (blockIdx.x * SCAN_WAVES + wave) * 16 + lo;
  const float alpha = __expf(log_alpha_p[0]);
  const float* wrow = Wh + (size_t)col * DIM;
  const unsigned nblk = gridDim.x;

  for (int t = 0; t < T_STEPS; ++t) {
    stage_frags(h + (size_t)t * BATCH * DIM, lds, 128);
    __syncthreads();

    float* wxb = outs + (size_t)t * BATCH * DIM;   // wx_t in, SiLU out
    v8f c0, c1, c2, c3;
#pragma unroll
    for (int i = 0; i < 8; ++i) {
      c0[i] = wxb[(size_t)(i + 8 * hi) * DIM + col];
      c1[i] = 0.f; c2[i] = 0.f; c3[i] = 0.f;
    }

    gemm_k(lds + (lane << 3), wrow, 2 * hi, c0, c1, c2, c3);

    float* hnext = h + (size_t)(t + 1) * BATCH * DIM;
#pragma unroll
    for (int i = 0; i < 8; ++i) {
      float pre = (c0[i] + c1[i]) + (c2[i] + c3[i]);
      int row = i + 8 * hi;                       // batch index
      float hp = lds_elem(lds, row, col);         // h_prev residual
      float hn = hp + alpha * tanhf(pre);
      float sig = 1.0f / (1.0f + __expf(-hn));
      hnext[(size_t)row * DIM + col] = hn;
      wxb[(size_t)row * DIM + col]   = hn * hn * sig;   // h * silu(h)
    }

    // device-wide barrier: monotone counter, target grows each step
    __threadfence();
    __syncthreads();
    if (threadIdx.x == 0) {
      atomicAdd((unsigned*)bar, 1u);
      const unsigned target = (unsigned)(t + 1) * nblk;
      while (*bar < target) { __builtin_amdgcn_s_sleep(2); }
    }
    __syncthreads();
    __threadfence();
  }
}

// ---------------------------------------------------------------------------
extern "C" void kernel_launch(void* const* d_in, const int* in_sizes, int n_in,
                              void* d_out, int out_size, void* d_ws, size_t ws_size,
                              hipStream_t stream) {
  const float* x         = (const float*)d_in[0];   // [T,B,D]
  const float* h0        = (const float*)d_in[1];   // [B,D]
  const float* Wx        = (const float*)d_in[2];   // [D,D]
  const float* Wh        = (const float*)d_in[3];   // [D,D]
  const float* bias      = (const float*)d_in[4];   // [D]
  const float* log_alpha = (const float*)d_in[5];   // [1]

  float* outs = (float*)d_out;                                  // [T,B,D]
  float* h    = outs + (size_t)T_STEPS * BATCH * DIM;           // [T+1,B,D]
  unsigned* bar = (unsigned*)d_ws;

  // h[0] = h0 (device-to-device, graph-capture safe)
  hipMemcpyAsync(h, h0, (size_t)BATCH * DIM * sizeof(float),
                 hipMemcpyDeviceToDevice, stream);

  // Phase 1: 2048 M-tiles x (8 col-groups of 8 waves) = full 32768x1024 GEMM
  dim3 g1((T_STEPS * BATCH) / 16, DIM / (8 * 16));
  e60_wx_gemm<<<g1, 256, 0, stream>>>(x, Wx, bias, outs, bar);

  // Phase 2: persistent scan, 16 WGs x 128 threads
  e60_scan<<<SCAN_WGS, SCAN_WAVES * 32, 0, stream>>>(Wh, log_alpha, outs, h, bar);
}